// RandLAUp_68496138437089
// MI455X (gfx1250) — compile-verified
//
#include <hip/hip_runtime.h>
#include <hip/hip_bf16.h>

// ---------------------------------------------------------------------------
// RandLA-style upsample: NN-interp + 1x1 conv + BN(train) + ReLU, MI455X.
// All matrix math via V_WMMA_F32_16X16X4_F32 (f32 end-to-end, matches ref).
// ---------------------------------------------------------------------------

typedef float v2f __attribute__((ext_vector_type(2)));
typedef float v8f __attribute__((ext_vector_type(8)));

constexpr int B  = 2;
constexpr int Nf = 16384;
constexpr int Nc = 4096;
constexpr int Cc = 256;
constexpr int Cf = 128;
constexpr int Co = 256;
constexpr int C  = Cc + Cf;           // 384
constexpr float BN_EPS = 1e-5f;

// ---------------------------------------------------------------------------
// Kernel 1: nearest coarse neighbor per fine point.
// One wave (32 thr) owns a 16-fine-point tile; loops coarse in 16-wide tiles.
// d2 = |xf|^2 - 2*(xf.xc) + |xc|^2 ; cross term via WMMA f32 16x16x4 (K=3 pad).
// All fragment builds are branchless (selects, not exec-masked loads).
// ---------------------------------------------------------------------------
__global__ __launch_bounds__(32) void nn_argmin_kernel(
    const float* __restrict__ xyz_fine,     // [B, Nf, 3]
    const float* __restrict__ xyz_coarse,   // [B, Nc, 3]
    int* __restrict__ idx_out)              // [B, Nf]
{
    const int tilesPerB = Nf / 16;
    const int b    = blockIdx.x / tilesPerB;
    const int i0   = (blockIdx.x % tilesPerB) * 16;
    const int lane = threadIdx.x;
    const int hf   = lane >> 4;      // half-wave: selects K pair / row group
    const int l    = lane & 15;

    __shared__ float fnorm[16];

    const float* xf = xyz_fine + (size_t)(b * Nf + i0) * 3;
    if (lane < 16) {
        float x = xf[lane * 3 + 0], y = xf[lane * 3 + 1], z = xf[lane * 3 + 2];
        fnorm[lane] = x * x + y * y + z * z;
    }
    __syncthreads();

    // A fragment (16x4): row = fine point l, K = (x,y | z,0) per half-wave.
    // Unconditional b96 load + branchless component select.
    v2f a;
    {
        const float* p = xf + l * 3;
        const float p0 = p[0], p1 = p[1], p2 = p[2];
        a.x = hf ? p2 : p0;
        a.y = hf ? 0.0f : p1;
    }

    float fn[8];
    #pragma unroll
    for (int r = 0; r < 8; ++r) fn[r] = fnorm[r + hf * 8];

    float best[8];
    int   bidx[8];
    #pragma unroll
    for (int r = 0; r < 8; ++r) { best[r] = 3.4e38f; bidx[r] = 0; }

    const float* xcb = xyz_coarse + (size_t)b * Nc * 3;
    #pragma unroll 2
    for (int j0 = 0; j0 < Nc; j0 += 16) {
        const float* p = xcb + (size_t)(j0 + l) * 3;
        const float p0 = p[0], p1 = p[1], p2 = p[2];   // one merged b96 load
        // B fragment (4x16): VGPR0 = K row {0|2}, VGPR1 = K row {1|3}, col = l.
        v2f bb;
        bb.x = hf ? p2 : p0;
        bb.y = hf ? 0.0f : p1;
        const float cn = p0 * p0 + p1 * p1 + p2 * p2;

        v8f dot = {};
        dot = __builtin_amdgcn_wmma_f32_16x16x4_f32(
            false, a, false, bb, (short)0, dot, false, false);

        #pragma unroll
        for (int r = 0; r < 8; ++r) {
            float d2 = fn[r] + cn - 2.0f * dot[r];
            if (d2 < best[r]) { best[r] = d2; bidx[r] = j0 + l; }
        }
    }

    // Row-wise argmin across the 16 lanes holding that row's columns.
    #pragma unroll
    for (int r = 0; r < 8; ++r) {
        float v = best[r];
        int   j = bidx[r];
        #pragma unroll
        for (int off = 8; off >= 1; off >>= 1) {
            float ov = __shfl_xor(v, off, 16);
            int   oj = __shfl_xor(j, off, 16);
            if (ov < v || (ov == v && oj < j)) { v = ov; j = oj; }
        }
        if (l == 0) idx_out[b * Nf + i0 + r + hf * 8] = j;
    }
}

// ---------------------------------------------------------------------------
// Kernel 2: gather + concat + 1x1 conv (GEMM) via WMMA f32 16x16x4.
// Block = 256 thr (8 waves) owns one 16-column N tile of one batch.
// Stage fused B tile [384 x 16] in LDS (24 KB); each wave does 2 M-tiles.
// Writes pre-BN y[B, Co, Nf] to d_out.
// ---------------------------------------------------------------------------
__global__ __launch_bounds__(256) void gemm_kernel(
    const float* __restrict__ feats_coarse, // [B, Cc, Nc]
    const float* __restrict__ feats_fine,   // [B, Cf, Nf]
    const float* __restrict__ W,            // [Co, C]
    const int*   __restrict__ idx,          // [B, Nf]
    float* __restrict__ y)                  // [B, Co, Nf]
{
    const int tilesPerB = Nf / 16;
    const int b  = blockIdx.x / tilesPerB;
    const int n0 = (blockIdx.x % tilesPerB) * 16;
    const int t  = threadIdx.x;

    __shared__ int   s_idx[16];
    __shared__ float Bs[C][16];   // fused tile, K-major: 384*16*4 = 24 KB

    if (t < 16) s_idx[t] = idx[b * Nf + n0 + t];
    __syncthreads();

    // Gather coarse rows: thread t = channel row t.
    {
        const float* fc = feats_coarse + (size_t)(b * Cc + t) * Nc;
        #pragma unroll
        for (int n = 0; n < 16; ++n) Bs[t][n] = fc[s_idx[n]];
    }
    // Fine rows (contiguous copy).
    if (t < Cf) {
        const float* ff = feats_fine + (size_t)(b * Cf + t) * Nf + n0;
        #pragma unroll
        for (int n = 0; n < 16; ++n) Bs[Cc + t][n] = ff[n];
    }
    __syncthreads();

    const int wave = t >> 5;
    const int lane = t & 31;
    const int hf   = lane >> 4;
    const int l    = lane & 15;

    #pragma unroll
    for (int mt = 0; mt < 2; ++mt) {
        const int m0 = (wave * 2 + mt) * 16;
        // A: W rows m0..m0+15; lane holds K pair (k0+2*hf, k0+2*hf+1) -> b64 load.
        const float* wrow = W + (size_t)(m0 + l) * C + 2 * hf;
        v8f acc = {};
        #pragma unroll 4
        for (int k0 = 0; k0 < C; k0 += 4) {
            v2f a = *(const v2f*)(wrow + k0);
            v2f bb;
            bb.x = Bs[k0 + 2 * hf][l];       // B VGPR0: K = k0 | k0+2
            bb.y = Bs[k0 + 2 * hf + 1][l];   // B VGPR1: K = k0+1 | k0+3
            acc = __builtin_amdgcn_wmma_f32_16x16x4_f32(
                false, a, false, bb, (short)0, acc, false, false);
        }
        #pragma unroll
        for (int r = 0; r < 8; ++r) {
            y[((size_t)(b * Co + m0 + r + hf * 8)) * Nf + n0 + l] = acc[r];
        }
    }
}

// ---------------------------------------------------------------------------
// Kernel 3: per-channel batch stats over (B, Nf), deterministic block reduce.
// Emits scale = gamma * rsqrt(var + eps), shift = beta - mean * scale.
// ---------------------------------------------------------------------------
__global__ __launch_bounds__(256) void stats_kernel(
    const float* __restrict__ y,
    const float* __restrict__ gamma,
    const float* __restrict__ beta,
    float* __restrict__ scale,
    float* __restrict__ shift)
{
    const int o = blockIdx.x;
    const int t = threadIdx.x;

    float s = 0.0f, s2 = 0.0f;
    for (int g = t; g < B * Nf; g += 256) {
        const int bb = g >> 14;          // g / Nf
        const int n  = g & (Nf - 1);
        const float v = y[((size_t)(bb * Co + o)) * Nf + n];
        s  += v;
        s2 += v * v;
    }

    __shared__ float ss[256];
    __shared__ float ss2[256];
    ss[t] = s; ss2[t] = s2;
    __syncthreads();
    for (int st = 128; st > 0; st >>= 1) {
        if (t < st) { ss[t] += ss[t + st]; ss2[t] += ss2[t + st]; }
        __syncthreads();
    }
    if (t == 0) {
        const float inv  = 1.0f / (float)(B * Nf);
        const float mean = ss[0] * inv;
        const float var  = ss2[0] * inv - mean * mean;   // biased var (jnp.var)
        const float sc   = gamma[o] * rsqrtf(var + BN_EPS);
        scale[o] = sc;
        shift[o] = beta[o] - mean * sc;
    }
}

// ---------------------------------------------------------------------------
// Kernel 4: in-place BN affine + ReLU over y[B, Co, Nf].
// ---------------------------------------------------------------------------
__global__ __launch_bounds__(256) void bn_relu_kernel(
    float* __restrict__ y,
    const float* __restrict__ scale,
    const float* __restrict__ shift)
{
    const size_t i = (size_t)blockIdx.x * 256 + threadIdx.x;
    const int o = (int)((i / (size_t)Nf) % (size_t)Co);
    const float v = y[i] * scale[o] + shift[o];
    y[i] = fmaxf(v, 0.0f);
}

// ---------------------------------------------------------------------------
extern "C" void kernel_launch(void* const* d_in, const int* in_sizes, int n_in,
                              void* d_out, int out_size, void* d_ws, size_t ws_size,
                              hipStream_t stream)
{
    const float* xyz_coarse   = (const float*)d_in[0];
    const float* feats_coarse = (const float*)d_in[1];
    const float* xyz_fine     = (const float*)d_in[2];
    const float* feats_fine   = (const float*)d_in[3];
    const float* W            = (const float*)d_in[4];
    const float* gamma        = (const float*)d_in[5];
    const float* beta         = (const float*)d_in[6];
    float* y = (float*)d_out;

    int*   idx   = (int*)d_ws;                                    // B*Nf ints
    float* scale = (float*)((char*)d_ws + (size_t)B * Nf * sizeof(int));
    float* shift = scale + Co;

    nn_argmin_kernel<<<B * (Nf / 16), 32, 0, stream>>>(xyz_fine, xyz_coarse, idx);
    gemm_kernel<<<B * (Nf / 16), 256, 0, stream>>>(feats_coarse, feats_fine, W, idx, y);
    stats_kernel<<<Co, 256, 0, stream>>>(y, gamma, beta, scale, shift);
    bn_relu_kernel<<<(B * Co * Nf) / 256, 256, 0, stream>>>(y, scale, shift);
}